// YOLOLoss_7060926234699
// MI455X (gfx1250) — compile-verified
//
#include <hip/hip_runtime.h>
#include <math.h>

#define HH   52
#define WW   52
#define HWSZ 2704            // 52*52
#define NMX  3
#define NCX  80
#define ATT  85
#define BSX  32
#define NGT  20
#define NGTT (BSX*NGT)       // 640
#define NTOT (BSX*NMX*HWSZ)  // 259584 = 1014 * 256

// anchors / 416
__device__ __constant__ float c_aw[9] = {10.f/416.f,16.f/416.f,33.f/416.f,30.f/416.f,62.f/416.f,
                                         59.f/416.f,116.f/416.f,156.f/416.f,373.f/416.f};
__device__ __constant__ float c_ah[9] = {13.f/416.f,30.f/416.f,23.f/416.f,61.f/416.f,45.f/416.f,
                                         119.f/416.f,90.f/416.f,198.f/416.f,326.f/416.f};

__device__ __forceinline__ float sigmoidf(float x) { return 1.f / (1.f + __expf(-x)); }

// ---------- Phase A: per-GT anchor assignment + corner/area precompute ----------
__global__ void yolo_pre(const float* __restrict__ gtb, const int* __restrict__ gtc,
                         int* __restrict__ info, float* __restrict__ boxes,
                         float* __restrict__ areas) {
    int idx = blockIdx.x * blockDim.x + threadIdx.x;
    if (idx >= NGTT) return;
    float gcx = gtb[idx*4+0], gcy = gtb[idx*4+1], gw = gtb[idx*4+2], gh = gtb[idx*4+3];

    // corner form + area, consumed by the hot per-cell loop
    float gx1 = gcx - 0.5f*gw, gy1 = gcy - 0.5f*gh;
    boxes[idx*4+0] = gx1;
    boxes[idx*4+1] = gy1;
    boxes[idx*4+2] = gx1 + gw;
    boxes[idx*4+3] = gy1 + gh;
    areas[idx]     = gw * gh;

    float ga = gw * gh;
    float best = -1.f; int bn = 0;
    for (int n = 0; n < 9; ++n) {
        float inter = fminf(gw, c_aw[n]) * fminf(gh, c_ah[n]);
        float uni   = ga + c_aw[n]*c_ah[n] - inter;
        float r     = inter / (uni + 1e-10f);
        if (r > best) { best = r; bn = n; }
    }
    int out = -1;
    if (bn < 3) {   // MASK = [0,1,2]
        int gi = (int)(gcx * (float)WW); gi = gi < 0 ? 0 : (gi > WW-1 ? WW-1 : gi);
        int gj = (int)(gcy * (float)HH); gj = gj < 0 ? 0 : (gj > HH-1 ? HH-1 : gj);
        int cls = gtc[idx];
        out = (bn << 22) | (cls << 14) | (gj << 7) | gi;  // non-negative
    }
    info[idx] = out;
}

// ---------- Phase B: per-cell main loss pass ----------
__global__ void __launch_bounds__(256)
yolo_main(const float* __restrict__ inp, const float* __restrict__ boxes,
          const float* __restrict__ areas, const int* __restrict__ info,
          float* __restrict__ acc) {
    __shared__ float s_box[NGTT*4];     // 10240 B (corner form)
    __shared__ float s_ga[NGTT];        //  2560 B (areas)
    __shared__ int   s_info[NGTT];      //  2560 B (packed targets)
    __shared__ float red[3];

    const int t = threadIdx.x;

    // --- CDNA5 async data-mover: stage GT corner/area/target tables into LDS ---
    unsigned lds_bx = (unsigned)(size_t)(void*)s_box;
    unsigned lds_ga = (unsigned)(size_t)(void*)s_ga;
    unsigned lds_if = (unsigned)(size_t)(void*)s_info;
    unsigned long long g_bx = (unsigned long long)(const void*)boxes;
    unsigned long long g_ga = (unsigned long long)(const void*)areas;
    unsigned long long g_if = (unsigned long long)(const void*)info;
    {
        unsigned la = lds_bx + (unsigned)(t * 16);
        unsigned long long ga = g_bx + (unsigned long long)(t * 16);
        asm volatile("global_load_async_to_lds_b128 %0, %1, off" :: "v"(la), "v"(ga) : "memory");
        la += 256u * 16u; ga += 256ull * 16ull;
        asm volatile("global_load_async_to_lds_b128 %0, %1, off" :: "v"(la), "v"(ga) : "memory");
    }
    if (t < 128) {  // remaining 2048 B of corner table (640 * 16B total)
        unsigned la = lds_bx + (unsigned)((512 + t) * 16);
        unsigned long long ga = g_bx + (unsigned long long)((512 + t) * 16);
        asm volatile("global_load_async_to_lds_b128 %0, %1, off" :: "v"(la), "v"(ga) : "memory");
    }
    if (t < 160) {  // 2560 B area table
        unsigned la = lds_ga + (unsigned)(t * 16);
        unsigned long long ga = g_ga + (unsigned long long)(t * 16);
        asm volatile("global_load_async_to_lds_b128 %0, %1, off" :: "v"(la), "v"(ga) : "memory");
    }
    if (t < 160) {  // 2560 B target table
        unsigned la = lds_if + (unsigned)(t * 16);
        unsigned long long ga = g_if + (unsigned long long)(t * 16);
        asm volatile("global_load_async_to_lds_b128 %0, %1, off" :: "v"(la), "v"(ga) : "memory");
    }
    if (t == 0) { red[0] = 0.f; red[1] = 0.f; red[2] = 0.f; }
    asm volatile("s_wait_asynccnt 0" ::: "memory");
    __syncthreads();

    const int n  = blockIdx.x * 256 + t;       // exactly NTOT threads total
    const int b  = n / (NMX * HWSZ);
    const int r0 = n - b * (NMX * HWSZ);
    const int a  = r0 / HWSZ;
    const int rc = r0 - a * HWSZ;
    const int j  = rc / WW;
    const int i  = rc - j * WW;

    const float* base = inp + (size_t)(b * (NMX*ATT) + a * ATT) * HWSZ + j * WW + i;
    float p0 = base[0];
    float p1 = base[HWSZ];
    float p2 = base[2*HWSZ];
    float p3 = base[3*HWSZ];
    float p4 = base[4*HWSZ];

    float sx = sigmoidf(p0), sy = sigmoidf(p1);
    float bw = __expf(p2) * c_aw[a], bh = __expf(p3) * c_ah[a];
    float cx = (sx + (float)i) * (1.f/(float)WW);
    float cy = (sy + (float)j) * (1.f/(float)HH);
    float px1 = cx - 0.5f*bw, py1 = cy - 0.5f*bh;
    float px2 = px1 + bw,     py2 = py1 + bh;
    float parea = bw * bh;

    // "low" test: best_iou only feeds (best_iou < 0.5), and
    // iou >= 0.5  <=>  2*inter >= union (union > 0) -> divide-free any-test
    bool high = false;
    const float* bx = &s_box[b * NGT * 4];
    const float* ar = &s_ga[b * NGT];
    #pragma unroll 4
    for (int g = 0; g < NGT; ++g) {
        float gx1 = bx[g*4+0], gy1 = bx[g*4+1], gx2 = bx[g*4+2], gy2 = bx[g*4+3];
        float iw = fminf(px2, gx2) - fmaxf(px1, gx1); iw = iw > 0.f ? iw : 0.f;
        float ih = fminf(py2, gy2) - fmaxf(py1, gy1); ih = ih > 0.f ? ih : 0.f;
        float inter = iw * ih;
        float denom = parea + ar[g] - inter + 1e-10f;
        high = high || (inter + inter >= denom);
    }

    // is this cell a scatter target? (last matching GT wins, like scatter-set)
    int tgt = -1;
    const int* inf = &s_info[b * NGT];
    for (int g = 0; g < NGT; ++g) {
        int v = inf[g];
        if (v >= 0) {
            int kk = v >> 22;
            int gj = (v >> 7) & 0x7f;
            int gi = v & 0x7f;
            if (kk == a && gj == j && gi == i) tgt = v;
        }
    }

    float conf = sigmoidf(p4);
    float l_sq = 0.f, l_w = 0.f;
    if (tgt >= 0) {
        float d = conf - 1.f;
        l_sq += d * d;
        l_w  += 1.f + (float)NCX;
        int cls = (tgt >> 14) & 0xff;
        const float te = 0.1f / (float)NCX;   // SMOOTH_EPS / NC
        for (int c = 0; c < NCX; ++c) {       // class channels only touched at target cells
            float s  = sigmoidf(base[(5 + c) * HWSZ]);
            float tv = te + (c == cls ? 0.9f : 0.f);
            float e  = s - tv;
            l_sq += e * e;
        }
    } else if (!high) {                       // best_iou < IGNORE_THR
        l_sq += conf * conf;
        l_w  += 1.f;
    } // else: t_obj == conf, w_obj == 0 -> zero contribution

    // wave32 shuffle reduction, then per-wave LDS atomics, then one global atomic/block
    float v0 = l_sq, v1 = l_w, v2 = conf;
    for (int off = 16; off > 0; off >>= 1) {
        v0 += __shfl_down(v0, off);
        v1 += __shfl_down(v1, off);
        v2 += __shfl_down(v2, off);
    }
    if ((t & 31) == 0) {
        atomicAdd(&red[0], v0);
        atomicAdd(&red[1], v1);
        atomicAdd(&red[2], v2);
    }
    __syncthreads();
    if (t == 0) {
        atomicAdd(&acc[0], red[0]);   // sum (out - tgt)^2 * w
        atomicAdd(&acc[1], red[1]);   // sum w
        atomicAdd(&acc[2], red[2]);   // total conf
    }
}

// ---------- Phase C: per-GT CIoU + stats ----------
__global__ void yolo_gt(const float* __restrict__ inp, const float* __restrict__ gtb,
                        const int* __restrict__ info, float* __restrict__ acc) {
    int idx = blockIdx.x * blockDim.x + threadIdx.x;
    if (idx >= NGTT) return;
    int v = info[idx];
    if (v < 0) return;                    // vf == 0 zeroes every contribution
    int b   = idx / NGT;
    int a   = v >> 22;
    int cls = (v >> 14) & 0xff;
    int j   = (v >> 7) & 0x7f;
    int i   = v & 0x7f;

    const float* base = inp + (size_t)(b * (NMX*ATT) + a * ATT) * HWSZ + j * WW + i;
    float sx = sigmoidf(base[0]),       sy = sigmoidf(base[HWSZ]);
    float bw = __expf(base[2*HWSZ]) * c_aw[a];
    float bh = __expf(base[3*HWSZ]) * c_ah[a];
    float conf = sigmoidf(base[4*HWSZ]);
    float clsp = sigmoidf(base[(5 + cls) * HWSZ]);
    float cx = (sx + (float)i) / (float)WW, cy = (sy + (float)j) / (float)HH;
    float px1 = cx - 0.5f*bw, py1 = cy - 0.5f*bh, px2 = px1 + bw, py2 = py1 + bh;

    float gcx = gtb[idx*4+0], gcy = gtb[idx*4+1], gw = gtb[idx*4+2], gh = gtb[idx*4+3];
    float gx1 = gcx - 0.5f*gw, gy1 = gcy - 0.5f*gh, gx2 = gx1 + gw, gy2 = gy1 + gh;

    const float eps = 1e-9f;
    float iw = fminf(px2, gx2) - fmaxf(px1, gx1); iw = iw > 0.f ? iw : 0.f;
    float ih = fminf(py2, gy2) - fmaxf(py1, gy1); ih = ih > 0.f ? ih : 0.f;
    float inter = iw * ih;
    float uni = gw*gh + bw*bh - inter;
    float iou = inter / (uni + eps);
    float cl = fminf(gx1, px1), ct = fminf(gy1, py1);
    float cr = fmaxf(gx2, px2), cb = fmaxf(gy2, py2);
    float c2 = (cr-cl)*(cr-cl) + (cb-ct)*(cb-ct) + eps;
    float dx = (gx1 + gx2 - px1 - px2) * 0.5f;
    float dy = (gy1 + gy2 - py1 - py2) * 0.5f;
    float d2 = dx*dx + dy*dy;
    float at = atanf(gw / (gh + eps)) - atanf(bw / (bh + eps));
    float vv = 0.40528473456f * at * at;              // 4/pi^2
    float alpha = vv / (1.f - iou + vv + eps);
    float ciou = iou - d2 / c2 - alpha * vv;

    float area  = gw * gh;
    float iou_w = 2.f - area;
    float om    = 1.f - ciou;

    atomicAdd(&acc[3], om * om * iou_w);              // CIoU loss numerator
    atomicAdd(&acc[4], iou_w);                        // CIoU loss weight sum
    atomicAdd(&acc[5], 1.f);                          // count
    atomicAdd(&acc[6], conf);                         // obj sum
    atomicAdd(&acc[7], iou > 0.5f ? 1.f : 0.f);       // recall sum
    atomicAdd(&acc[8], iou);                          // iou sum
    atomicAdd(&acc[9], clsp);                         // class prob sum
}

// ---------- Phase D: finalize 7 scalars ----------
__global__ void yolo_fin(const float* __restrict__ acc, float* __restrict__ out) {
    float loss_main = acc[0] / acc[1];
    float count = acc[5];
    float cnt = fmaxf(count, 1.f);
    float iou_loss = acc[3] / fmaxf(acc[4], 1e-10f) / cnt;
    out[0] = loss_main + 0.01f * iou_loss;              // loss
    out[1] = acc[7] / cnt;                              // recall
    out[2] = acc[8] / cnt;                              // avg_iou
    out[3] = acc[6] / cnt;                              // obj_avg
    out[4] = (acc[2] - acc[6]) / ((float)NTOT - count); // no_obj
    out[5] = acc[9] / cnt;                              // cls_avg
    out[6] = count / (float)BSX;                        // count / bs
}

extern "C" void kernel_launch(void* const* d_in, const int* in_sizes, int n_in,
                              void* d_out, int out_size, void* d_ws, size_t ws_size,
                              hipStream_t stream) {
    (void)in_sizes; (void)n_in; (void)out_size; (void)ws_size;
    const float* inp = (const float*)d_in[0];
    const float* gtb = (const float*)d_in[1];
    const int*   gtc = (const int*)d_in[2];
    float* out   = (float*)d_out;
    float* acc   = (float*)d_ws;                     // 16 floats of accumulators @ 0
    int*   info  = (int*)((char*)d_ws + 64);         // 640 packed targets   @ 64   (2560 B)
    float* boxes = (float*)((char*)d_ws + 2624);     // 640x4 corner boxes   @ 2624 (10240 B)
    float* areas = (float*)((char*)d_ws + 12864);    // 640 areas            @ 12864 (2560 B)

    hipMemsetAsync(d_ws, 0, 64, stream);             // re-zero accumulators every call
    yolo_pre<<<2, 320, 0, stream>>>(gtb, gtc, info, boxes, areas);
    yolo_main<<<NTOT / 256, 256, 0, stream>>>(inp, boxes, areas, info, acc);
    yolo_gt<<<2, 320, 0, stream>>>(inp, gtb, info, acc);
    yolo_fin<<<1, 1, 0, stream>>>(acc, out);
}